// MultiHeadDotProductAttention_88656714925388
// MI455X (gfx1250) — compile-verified
//
#include <hip/hip_runtime.h>

// ---------------- problem constants (from reference) ----------------
// b=16, n=577, c=768, h=12, d=64, keep_rate=0.7 -> left_tokens = ceil(0.7*576) = 404
#define BB   16
#define NN   577
#define CC   768
#define HH   12
#define DD   64
#define LT   404
#define N1   576            // n-1
#define SCALE 0.125f        // (c/h)^-0.5 = 1/8

// output regions (flat float32, in return order: x, index, idx, cls_attn)
#define X_ELEMS      (BB * NN * CC)        // 7,090,176
#define INDEX_ELEMS  (BB * LT * CC)        // 4,964,352
#define IDX_ELEMS    (BB * LT)             // 6,464
#define CLS_ELEMS    (BB * N1)             // 9,216

// workspace layout (floats)
#define WS_Q0    0                         // [16][768]
#define WS_M     (WS_Q0 + BB * CC)         // [16][768][12]
#define WS_KB    (WS_M + BB * CC * HH)     // [16][12]
#define WS_DOTS  (WS_KB + BB * HH)         // [16][12][577]
#define WS_TOTAL (WS_DOTS + BB * HH * NN)  // ~270k floats (~1.06 MB)

typedef __attribute__((ext_vector_type(2))) float v2f;
typedef __attribute__((ext_vector_type(8))) float v8f;

// ---------------------------------------------------------------
// 1) x passthrough: pure bandwidth, float4 grid-stride copy
// ---------------------------------------------------------------
__global__ void copy_x_kernel(const float4* __restrict__ src, float4* __restrict__ dst, int n4) {
    for (int i = blockIdx.x * blockDim.x + threadIdx.x; i < n4; i += gridDim.x * blockDim.x)
        dst[i] = src[i];
}

// ---------------------------------------------------------------
// 2) q0[b, c2] = x[b, 0, :] @ Wq[:, c2] + bq[c2]    (16 x 768, K = 768)
//    fp32 WMMA 16x16x4: M = batch (16), one wave per 16-wide N tile.
//    A 16x4 layout: lanes 0-15 hold M=lane, VGPR0/1 = K 2*khalf / 2*khalf+1
//    B 4x16 layout: lanes 0-15 hold N=lane, VGPR0/1 = K 2*khalf / 2*khalf+1
// ---------------------------------------------------------------
__global__ void q0_wmma_kernel(const float* __restrict__ x,
                               const float* __restrict__ wqkv,
                               const float* __restrict__ bqkv,
                               float* __restrict__ q0) {
    const int n0    = blockIdx.x * 16;      // output column tile in [0, 768)
    const int lane  = threadIdx.x & 31;
    const int l15   = lane & 15;
    const int khalf = lane >> 4;            // 0 or 1
    const size_t xrow = (size_t)l15 * (size_t)(NN * CC);  // token 0 of batch l15
    v8f acc = {};
    for (int k0 = 0; k0 < CC; k0 += 4) {
        const int ka = k0 + 2 * khalf;
        v2f a, bmat;
        a.x = x[xrow + ka];
        a.y = x[xrow + ka + 1];
        bmat.x = wqkv[(size_t)ka       * (3 * CC) + n0 + l15];   // q part: cols [0,768)
        bmat.y = wqkv[(size_t)(ka + 1) * (3 * CC) + n0 + l15];
        acc = __builtin_amdgcn_wmma_f32_16x16x4_f32(false, a, false, bmat,
                                                    (short)0, acc, false, false);
    }
    const float bias = bqkv[n0 + l15];
    #pragma unroll
    for (int r = 0; r < 8; ++r) {
        const int row = r + 8 * khalf;              // batch index
        q0[row * CC + n0 + l15] = acc[r] + bias;
    }
}

// ---------------------------------------------------------------
// 3) fold K-weights through q0:
//    m[b, c1, h] = sum_d WK[c1, h*64+d] * q0[b, h*64+d]
//    (then dots0[b,h,j] = x[b,j,:] . m[b,:,h] — avoids materializing k)
// ---------------------------------------------------------------
__global__ void fold_m_kernel(const float* __restrict__ wqkv,
                              const float* __restrict__ q0,
                              float* __restrict__ mfold) {
    const int t = blockIdx.x * blockDim.x + threadIdx.x;   // (b, c1)
    if (t >= BB * CC) return;
    const int b = t / CC, c1 = t % CC;
    const float* __restrict__ wrow = wqkv + (size_t)c1 * (3 * CC) + CC;  // k part of row c1
    const float* __restrict__ qb   = q0 + b * CC;
    #pragma unroll 1
    for (int h = 0; h < HH; ++h) {
        float s = 0.f;
        #pragma unroll 8
        for (int d = 0; d < DD; ++d)
            s = fmaf(wrow[h * DD + d], qb[h * DD + d], s);
        mfold[(size_t)t * HH + h] = s;
    }
}

// k-bias contribution: kb[b,h] = sum_d q0[b,h*64+d] * bk[h*64+d]
__global__ void kbias_kernel(const float* __restrict__ q0,
                             const float* __restrict__ bqkv,
                             float* __restrict__ kb) {
    const int t = threadIdx.x;
    if (t >= BB * HH) return;
    const int b = t / HH, h = t % HH;
    float s = 0.f;
    for (int d = 0; d < DD; ++d)
        s = fmaf(q0[b * CC + h * DD + d], bqkv[CC + h * DD + d], s);
    kb[t] = s;
}

// ---------------------------------------------------------------
// 4) dots0[b, h, j] = scale * (x[b,j,:] . m[b,:,h] + kb[b,h])
//    per-batch GEMM M=577 (pad 592) x N=12 (pad 16), K=768, fp32 WMMA.
//    grid = (37 M-tiles, 16 batches), one wave each.
// ---------------------------------------------------------------
__global__ void dots_wmma_kernel(const float* __restrict__ x,
                                 const float* __restrict__ mfold,
                                 const float* __restrict__ kb,
                                 float* __restrict__ dots) {
    const int b     = blockIdx.y;
    const int j0    = blockIdx.x * 16;
    const int lane  = threadIdx.x & 31;
    const int l15   = lane & 15;
    const int khalf = lane >> 4;
    int j = j0 + l15; if (j > NN - 1) j = NN - 1;          // clamp loads, keep EXEC full
    const float* __restrict__ arow = x + ((size_t)b * NN + j) * CC;
    const float* __restrict__ mb   = mfold + (size_t)b * CC * HH;
    const bool hvalid = (l15 < HH);
    v8f acc = {};
    for (int k0 = 0; k0 < CC; k0 += 4) {
        const int ka = k0 + 2 * khalf;
        v2f a, bmat;
        a.x = arow[ka];
        a.y = arow[ka + 1];
        bmat.x = hvalid ? mb[(size_t)ka * HH + l15]       : 0.f;
        bmat.y = hvalid ? mb[(size_t)(ka + 1) * HH + l15] : 0.f;
        acc = __builtin_amdgcn_wmma_f32_16x16x4_f32(false, a, false, bmat,
                                                    (short)0, acc, false, false);
    }
    if (hvalid) {
        const int h = l15;
        const float kbb = kb[b * HH + h];
        #pragma unroll
        for (int r = 0; r < 8; ++r) {
            const int jj = j0 + r + 8 * khalf;             // token row
            if (jj < NN)
                dots[((size_t)b * HH + h) * NN + jj] = SCALE * (acc[r] + kbb);
        }
    }
}

// ---------------------------------------------------------------
// 5) softmax over the HEADS axis (reference quirk), then mean over heads:
//    cls_attn[b, j-1] = (1/12) * sum_h exp(d_h - max) / sum_h' exp(d_h' - max)
//    (analytically == 1/12; computed faithfully so top-k tie structure matches)
// ---------------------------------------------------------------
__global__ void cls_softmax_kernel(const float* __restrict__ dots,
                                   float* __restrict__ cls_out) {
    const int t = blockIdx.x * blockDim.x + threadIdx.x;
    if (t >= BB * N1) return;
    const int b = t / N1, j = (t % N1) + 1;   // skip CLS column
    float v[HH], mx = -3.4e38f;
    #pragma unroll
    for (int h = 0; h < HH; ++h) {
        v[h] = dots[((size_t)b * HH + h) * NN + j];
        mx = fmaxf(mx, v[h]);
    }
    float s = 0.f;
    #pragma unroll
    for (int h = 0; h < HH; ++h) { v[h] = __expf(v[h] - mx); s += v[h]; }
    float acc = 0.f;
    #pragma unroll
    for (int h = 0; h < HH; ++h) acc += v[h] / s;          // per-element div like jax.nn.softmax
    cls_out[t] = acc * (1.0f / HH);
}

// ---------------------------------------------------------------
// 6) stable top-k (k=404 of 576) + ascending index sort, per batch.
//    rank = #{i : v_i > v_t or (v_i == v_t and i < t)}  (jax stable tie-break)
//    selected iff rank < 404; output position = prefix count of selected.
// ---------------------------------------------------------------
__global__ void topk_kernel(const float* __restrict__ cls,
                            float* __restrict__ out_idx) {
    __shared__ float sv[N1];
    __shared__ int   sel[N1];
    const int b = blockIdx.x, t = threadIdx.x;   // blockDim.x == 576
    const float v = cls[b * N1 + t];
    sv[t] = v;
    __syncthreads();
    int rank = 0;
    for (int i = 0; i < N1; ++i) {
        const float vi = sv[i];
        rank += (vi > v) || (vi == v && i < t);
    }
    sel[t] = (rank < LT);
    __syncthreads();
    if (rank < LT) {
        int pos = 0;
        for (int i = 0; i < t; ++i) pos += sel[i];
        out_idx[b * LT + pos] = (float)t;
    }
}

// ---------------------------------------------------------------
// 7) index = broadcast idx[b,t] over the channel dim -> [16, 404, 768]
// ---------------------------------------------------------------
__global__ void bcast_index_kernel(const float* __restrict__ idx,
                                   float4* __restrict__ out_index) {
    const int total4 = INDEX_ELEMS / 4;
    for (int i = blockIdx.x * blockDim.x + threadIdx.x; i < total4; i += gridDim.x * blockDim.x) {
        const int bt = (i * 4) / CC;            // (b*404 + t); CC % 4 == 0 keeps rows aligned
        const float v = idx[bt];
        out_index[i] = make_float4(v, v, v, v);
    }
}

// ---------------------------------------------------------------
extern "C" void kernel_launch(void* const* d_in, const int* in_sizes, int n_in,
                              void* d_out, int out_size, void* d_ws, size_t ws_size,
                              hipStream_t stream) {
    const float* x     = (const float*)d_in[0];
    const float* wqkv  = (const float*)d_in[1];
    const float* bqkv  = (const float*)d_in[2];
    // d_in[3] (w_out), d_in[4] (b_out): dead in the returned outputs
    float* out = (float*)d_out;
    float* ws  = (float*)d_ws;

    float* q0    = ws + WS_Q0;
    float* mfold = ws + WS_M;
    float* kb    = ws + WS_KB;
    float* dots  = ws + WS_DOTS;

    float* out_x     = out;
    float* out_index = out + X_ELEMS;
    float* out_idx   = out + X_ELEMS + INDEX_ELEMS;
    float* out_cls   = out + X_ELEMS + INDEX_ELEMS + IDX_ELEMS;

    // 1) x passthrough (dominant cost: ~57 MB of traffic)
    copy_x_kernel<<<4096, 256, 0, stream>>>((const float4*)x, (float4*)out_x, X_ELEMS / 4);

    // 2) q0 = x[:,0,:] @ Wq + bq   (WMMA fp32)
    q0_wmma_kernel<<<CC / 16, 32, 0, stream>>>(x, wqkv, bqkv, q0);

    // 3) fold Wk through q0; k-bias term
    fold_m_kernel<<<(BB * CC + 255) / 256, 256, 0, stream>>>(wqkv, q0, mfold);
    kbias_kernel<<<1, BB * HH, 0, stream>>>(q0, bqkv, kb);

    // 4) dots0 = scale * (x[b] @ m[b] + kb)   (WMMA fp32, 37x16 waves)
    dots_wmma_kernel<<<dim3((NN + 15) / 16, BB), 32, 0, stream>>>(x, mfold, kb, dots);

    // 5) softmax over heads -> cls_attn (written straight to its output region)
    cls_softmax_kernel<<<(BB * N1 + 255) / 256, 256, 0, stream>>>(dots, out_cls);

    // 6) stable top-404 + ascending order -> idx
    topk_kernel<<<BB, N1, 0, stream>>>(out_cls, out_idx);

    // 7) broadcast idx over channels -> index
    bcast_index_kernel<<<4096, 256, 0, stream>>>(out_idx, (float4*)out_index);
}